// AttentionPooling_9234179686675
// MI455X (gfx1250) — compile-verified
//
#include <hip/hip_runtime.h>
#include <hip/hip_bf16.h>

typedef float v2f __attribute__((ext_vector_type(2)));
typedef float v8f __attribute__((ext_vector_type(8)));

#define DCONST 256

// ---------------- fill (bit pattern) ----------------
__global__ void fill_u32(unsigned int* __restrict__ p, long long n, unsigned int v) {
    long long i = (long long)blockIdx.x * blockDim.x + threadIdx.x;
    long long stride = (long long)gridDim.x * blockDim.x;
    for (; i < n; i += stride) p[i] = v;
}

// ---------------- gate logits: g[n] = fea[n,:]·Wg + bg ----------------
__global__ __launch_bounds__(256)
void gate_kernel(const float* __restrict__ fea, const float* __restrict__ Wg,
                 const float* __restrict__ bgp, float* __restrict__ g, int N) {
    int wv = (int)((blockIdx.x * 256 + threadIdx.x) >> 5);
    int lane = threadIdx.x & 31;
    if (wv >= N) return;
    const float4* fr = (const float4*)(fea + (size_t)wv * DCONST) + (lane << 1);
    const float4* wr = (const float4*)Wg + (lane << 1);
    float4 a0 = fr[0], a1 = fr[1];
    float4 w0 = wr[0], w1 = wr[1];
    float s = a0.x*w0.x + a0.y*w0.y + a0.z*w0.z + a0.w*w0.w
            + a1.x*w1.x + a1.y*w1.y + a1.z*w1.z + a1.w*w1.w;
    s += __shfl_xor(s, 16);
    s += __shfl_xor(s, 8);
    s += __shfl_xor(s, 4);
    s += __shfl_xor(s, 2);
    s += __shfl_xor(s, 1);
    if (lane == 0) g[wv] = s + bgp[0];
}

// ---------------- segment max (sign-aware float atomic max) ----------------
__device__ __forceinline__ void atomicMaxF(float* addr, float v) {
    if (v >= 0.0f) atomicMax((int*)addr, __float_as_int(v));
    else           atomicMin((unsigned int*)addr, __float_as_uint(v));
}

__global__ __launch_bounds__(256)
void segmax_kernel(const float* __restrict__ g, const int* __restrict__ index,
                   float* __restrict__ smax, int N) {
    int i = blockIdx.x * 256 + threadIdx.x;
    if (i < N) atomicMaxF(&smax[index[i]], g[i]);
}

// ---------------- weighted pooling: pooled[s,:] += e_n * fea[n,:]; gsum[s] += e_n ----
#define ROWS_PER_WAVE 16
__global__ __launch_bounds__(256)
void pool_kernel(const float* __restrict__ fea, const float* __restrict__ g,
                 const int* __restrict__ index, const float* __restrict__ smax,
                 float* __restrict__ gsum, float* __restrict__ pooled, int N) {
    int wv = (int)((blockIdx.x * 256 + threadIdx.x) >> 5);
    int lane = threadIdx.x & 31;
    long long row0 = (long long)wv * ROWS_PER_WAVE;
    if (row0 >= N) return;
    long long rend = row0 + ROWS_PER_WAVE;
    if (rend > N) rend = N;

    float acc[8];
#pragma unroll
    for (int j = 0; j < 8; ++j) acc[j] = 0.0f;

    int cur = index[row0];
    for (long long r = row0; r < rend; ++r) {
        int seg = index[r];
        if (seg != cur) {
            float* pp = pooled + (size_t)cur * DCONST + (lane << 3);
#pragma unroll
            for (int j = 0; j < 8; ++j) { atomicAdd(pp + j, acc[j]); acc[j] = 0.0f; }
            cur = seg;
        }
        float w = __expf(g[r] - smax[seg]);
        if (lane == 0) atomicAdd(&gsum[seg], w);
        const float4* fr = (const float4*)(fea + (size_t)r * DCONST) + (lane << 1);
        float4 a0 = fr[0], a1 = fr[1];
        acc[0] += w * a0.x; acc[1] += w * a0.y; acc[2] += w * a0.z; acc[3] += w * a0.w;
        acc[4] += w * a1.x; acc[5] += w * a1.y; acc[6] += w * a1.z; acc[7] += w * a1.w;
    }
    float* pp = pooled + (size_t)cur * DCONST + (lane << 3);
#pragma unroll
    for (int j = 0; j < 8; ++j) atomicAdd(pp + j, acc[j]);
}

// ---------------- WMMA fp32 GEMM: out = (pooled/(gsum+eps)) @ Wm + bm*(gsum/(gsum+eps)) ----
// block = 128 threads = 4 full waves; each block computes a 16x256 strip of out,
// each wave a 16x64 strip (4 accumulator tiles), K stepped by 4 via V_WMMA_F32_16X16X4_F32.
__global__ __launch_bounds__(128)
void attn_gemm_wmma(const float* __restrict__ pooled, const float* __restrict__ gsum,
                    const float* __restrict__ Wm, const float* __restrict__ bm,
                    float* __restrict__ out, int S) {
    const int lane = threadIdx.x & 31;
    const int wv   = threadIdx.x >> 5;       // 0..3
    const int row0 = blockIdx.x << 4;        // 16 rows / block
    const int ml   = lane & 15;              // M (for A) / N (for B,D) within tile
    const int kh   = (lane >> 4) << 1;       // K sub-offset: 0 (lanes 0-15) or 2 (lanes 16-31)
    const int n0   = wv << 6;                // 64-wide N strip per wave

    int arow = row0 + ml;
    int arowc = (arow < S) ? arow : (S - 1);
    float inv = (arow < S) ? (1.0f / (gsum[arowc] + 1e-10f)) : 0.0f;

    const float* aBase = pooled + (size_t)arowc * DCONST + kh;

    v8f c0 = {0.f,0.f,0.f,0.f,0.f,0.f,0.f,0.f};
    v8f c1 = c0, c2 = c0, c3 = c0;

    for (int k = 0; k < DCONST; k += 4) {
        // A fragment 16x4: row = arow, cols k+kh, k+kh+1 ; normalize on load
        const float* ap = aBase + k;
        v2f a; a.x = ap[0] * inv; a.y = ap[1] * inv;
        // B fragments 4x16: rows k+kh, k+kh+1 ; col = n0 + t*16 + ml
        const float* bp = Wm + (size_t)(k + kh) * DCONST + ml;
        v2f b0; b0.x = bp[n0 +  0]; b0.y = bp[n0 +  0 + DCONST];
        v2f b1; b1.x = bp[n0 + 16]; b1.y = bp[n0 + 16 + DCONST];
        v2f b2; b2.x = bp[n0 + 32]; b2.y = bp[n0 + 32 + DCONST];
        v2f b3; b3.x = bp[n0 + 48]; b3.y = bp[n0 + 48 + DCONST];

        c0 = __builtin_amdgcn_wmma_f32_16x16x4_f32(false, a, false, b0, (short)0, c0, false, false);
        c1 = __builtin_amdgcn_wmma_f32_16x16x4_f32(false, a, false, b1, (short)0, c1, false, false);
        c2 = __builtin_amdgcn_wmma_f32_16x16x4_f32(false, a, false, b2, (short)0, c2, false, false);
        c3 = __builtin_amdgcn_wmma_f32_16x16x4_f32(false, a, false, b3, (short)0, c3, false, false);
    }

    // epilogue: D tile layout — VGPR r: lanes 0-15 -> M=row0+r, lanes 16-31 -> M=row0+8+r; N=ml
    float bmv0 = bm[n0 +  0 + ml];
    float bmv1 = bm[n0 + 16 + ml];
    float bmv2 = bm[n0 + 32 + ml];
    float bmv3 = bm[n0 + 48 + ml];
    int mBase = row0 + ((lane >> 4) << 3);
#pragma unroll
    for (int r = 0; r < 8; ++r) {
        int m = mBase + r;
        if (m < S) {
            float gs = gsum[m];
            float wt = gs / (gs + 1e-10f);       // sum of normalized gates (exact ref epsilon)
            float* orow = out + (size_t)m * DCONST;
            orow[n0 +  0 + ml] = c0[r] + bmv0 * wt;
            orow[n0 + 16 + ml] = c1[r] + bmv1 * wt;
            orow[n0 + 32 + ml] = c2[r] + bmv2 * wt;
            orow[n0 + 48 + ml] = c3[r] + bmv3 * wt;
        }
    }
}

// ---------------- host ----------------
extern "C" void kernel_launch(void* const* d_in, const int* in_sizes, int n_in,
                              void* d_out, int out_size, void* d_ws, size_t ws_size,
                              hipStream_t stream) {
    const float* fea = (const float*)d_in[0];
    const float* Wg  = (const float*)d_in[1];
    const float* bg  = (const float*)d_in[2];
    const float* Wm  = (const float*)d_in[3];
    const float* bm  = (const float*)d_in[4];
    const int*   idx = (const int*)d_in[5];

    const int D = in_sizes[4];            // 256
    const int N = in_sizes[5];            // 500000
    const int S = out_size / D;           // 50000
    float* out = (float*)d_out;

    // workspace carve-out
    char* ws = (char*)d_ws;
    size_t off = 0;
    auto al = [](size_t x) { return (x + 255) & ~(size_t)255; };
    float* g      = (float*)(ws + off); off += al((size_t)N * 4);
    float* smax   = (float*)(ws + off); off += al((size_t)S * 4);
    float* gsum   = (float*)(ws + off); off += al((size_t)S * 4);
    float* pooled = (float*)(ws + off); off += al((size_t)S * D * 4);
    (void)ws_size; (void)n_in;

    // 1) init workspace: smax = -inf, gsum = 0, pooled = 0
    {
        long long nS = S;
        int blk = (int)((nS + 255) / 256);
        fill_u32<<<blk, 256, 0, stream>>>((unsigned int*)smax, nS, 0xFF800000u);
        fill_u32<<<blk, 256, 0, stream>>>((unsigned int*)gsum, nS, 0u);
        long long nP = (long long)S * D;
        int blkP = (int)((nP + 1023) / 1024);
        fill_u32<<<blkP, 256, 0, stream>>>((unsigned int*)pooled, nP, 0u);
    }

    // 2) gate logits (one wave per row; 8 waves per block)
    {
        int blocks = (N + 7) / 8;
        gate_kernel<<<blocks, 256, 0, stream>>>(fea, Wg, bg, g, N);
    }

    // 3) segment max
    {
        int blocks = (N + 255) / 256;
        segmax_kernel<<<blocks, 256, 0, stream>>>(g, idx, smax, N);
    }

    // 4) weighted pooling (unnormalized) + gsum
    {
        long long waves = ((long long)N + ROWS_PER_WAVE - 1) / ROWS_PER_WAVE;
        int blocks = (int)((waves + 7) / 8);
        pool_kernel<<<blocks, 256, 0, stream>>>(fea, g, idx, smax, gsum, pooled, N);
    }

    // 5) WMMA fp32 GEMM + bias epilogue
    {
        int blocks = (S + 15) / 16;
        attn_gemm_wmma<<<blocks, 128, 0, stream>>>(pooled, gsum, Wm, bm, out, S);
    }
}